// SHM_8667244003397
// MI455X (gfx1250) — compile-verified
//
#include <hip/hip_runtime.h>
#include <cstdint>

// ---------------- configuration ----------------
#define WAVES_PER_BLOCK 8
#define SLOTS 32                 // ring depth (timesteps buffered per wave)
#define SLOT_F 272               // floats per slot: 128 (b0) + 16 pad + 128 (b1)
#define KAHEAD (SLOTS - 1)       // issue distance = 31 -> max 62 outstanding async ops (ASYNCcnt max 63)
#define PF_DIST 192              // global_prefetch distance (timesteps)
#define KLU_C 0.9f

// ---------------- fast device math ----------------
__device__ __forceinline__ float fast_rcp(float x) { return __builtin_amdgcn_rcpf(x); }

__device__ __forceinline__ float fast_sigmoid(float x) {
#if __has_builtin(__builtin_amdgcn_tanhf)
  // sigmoid(x) = 0.5 * (1 + tanh(x/2))  -> single v_tanh_f32 (gfx1250 trans op)
  return fmaf(0.5f, __builtin_amdgcn_tanhf(0.5f * x), 0.5f);
#else
  // 1 / (1 + exp(-x)) via v_exp_f32 (base-2) + v_rcp_f32
  return __builtin_amdgcn_rcpf(1.0f + __builtin_amdgcn_exp2f(-1.4426950408889634f * x));
#endif
}

// psi = r^beta with r in [0,1]; r==0 -> log2 gives -inf -> exp2 gives 0 (correct: 0^beta, beta>=1)
__device__ __forceinline__ float fast_pow(float r, float beta) {
  return __builtin_amdgcn_exp2f(beta * __builtin_amdgcn_logf(r));
}

// ---------------- CDNA5 async copy: global -> LDS, 2x b128 (one per basin) ----------------
// GVS addressing: saddr = uniform base (SGPR pair), vaddr = per-lane 32-bit byte offset,
// vdst = per-lane LDS byte address. Tracked with ASYNCcnt, completes in order.
__device__ __forceinline__ void async_copy_slot(const float* base, unsigned g0, unsigned g1,
                                                unsigned l0, unsigned l1) {
  asm volatile(
      "global_load_async_to_lds_b128 %0, %1, %4\n\t"
      "global_load_async_to_lds_b128 %2, %3, %4"
      :
      : "v"(l0), "v"(g0), "v"(l1), "v"(g1), "s"(base)
      : "memory");
}

// steady-state: outstanding <= 60 ops (30 pairs) => oldest pair (slot t) has landed
#define WAIT_ASYNC_STEADY() asm volatile("s_wait_asynccnt 60" ::: "memory")
#define WAIT_ASYNC_ZERO()   asm volatile("s_wait_asynccnt 0"  ::: "memory")

// ---------------- kernel ----------------
__global__ __launch_bounds__(256, 1) void shm_scan_kernel(
    const float* __restrict__ xc,     // [B, T, 3]
    const float* __restrict__ lstm,   // [B, T, 128]  (8 params x 16 models)
    float* __restrict__ out,          // [B, T]
    int T) {
  __shared__ float smem[WAVES_PER_BLOCK * SLOTS * SLOT_F];  // 278,528 bytes (<= 320 KB WGP LDS)

  const int tid  = (int)threadIdx.x;
  const int lane = tid & 31;
  const int wv   = tid >> 5;
  const int half = lane >> 4;       // 0 -> basin b0, 1 -> basin b0+1
  const int mm   = lane & 15;       // model index within basin
  const int gw   = (int)blockIdx.x * WAVES_PER_BLOCK + wv;
  const int b0   = gw * 2;
  const int b    = b0 + half;

  // LDS byte addresses (truncate generic pointer -> 32-bit LDS offset)
  const unsigned lds_wave  = (unsigned)(uintptr_t)(&smem[wv * SLOTS * SLOT_F]);
  const unsigned lane16    = (unsigned)lane * 16u;

  // global byte offsets into lstm for async b128 streaming (512 B per (b,t) row)
  const unsigned rowB      = 512u;
  unsigned       goff0     = (unsigned)(b0 * T) * rowB + lane16;
  const unsigned db1       = (unsigned)T * rowB;

  // ---- prologue: fill KAHEAD ring slots ----
#pragma unroll 1
  for (int s = 0; s < KAHEAD; ++s) {
    const unsigned l0 = lds_wave + (unsigned)s * (SLOT_F * 4u) + lane16;
    async_copy_slot(lstm, goff0, goff0 + db1, l0, l0 + 576u);
    goff0 += rowB;
  }

  // states
  float ss = 0.0f, sf = 1.0f, su = 5.0f, si = 10.0f, sb = 15.0f;

  const float* xr   = xc + (size_t)b * (size_t)T * 3u;  // forcing row for this lane's basin
  const size_t outb = (size_t)b * (size_t)T;
  const size_t lrow = (size_t)b * (size_t)T;            // in (b,t) rows

  const float LO[8]  = {0.0f, 10.0f, 20.0f, 1.0f, 0.0f, 1.0f, 1.0f, 10.0f};
  const float RNG[8] = {10.0f, 50.0f, 680.0f, 5.0f, 1.0f, 19.0f, 99.0f, 990.0f};

  auto step = [&](int t) {
    const int slot = t & (SLOTS - 1);
    const float* sp = &smem[wv * SLOTS * SLOT_F + slot * SLOT_F + half * 144 + mm];

    // raw lstm values for this (b,t,m): i-major, stride 16 floats (bank-conflict-free:
    // the 64B inter-stripe pad puts lanes 0-15 and 16-31 on disjoint bank groups)
    float raw[8];
#pragma unroll
    for (int i = 0; i < 8; ++i) raw[i] = sp[i * 16];

    // forcing (broadcast across the 16-lane group; kept hot via prefetch)
    const float prec = xr[3 * t + 0];
    const float pet  = xr[3 * t + 1];
    const float temp = xr[3 * t + 2];

    // prefetch lstm ~PF_DIST steps ahead into cache (gfx1250 global_prefetch_b8)
    {
      int pt = t + PF_DIST; if (pt >= T) pt = T - 1;
      const char* pfa = (const char*)lstm + ((lrow + (size_t)pt) * 512u) + (size_t)mm * 32u;
      __builtin_prefetch(pfa, 0, 0);
      __builtin_prefetch((const char*)(xr + 3 * t) + 512, 0, 0);
    }

    // sigmoid-rescaled dynamic parameters
    float pr[8];
#pragma unroll
    for (int i = 0; i < 8; ++i) pr[i] = fmaf(RNG[i], fast_sigmoid(raw[i]), LO[i]);
    const float dd = pr[0], f_thr = pr[1], sumax = pr[2], beta = pr[3];
    const float perc = pr[4], kf = pr[5], ki = pr[6], kb = pr[7];
    const float inv_sumax = fast_rcp(sumax);
    const float pwp = 0.8f * sumax;

    // forcing split (snow branch)
    const bool  frozen = temp < 0.0f;
    const float sm = frozen ? 0.0f : temp * dd;
    const float lp = frozen ? 0.0f : prec;
    const float sn = frozen ? prec : 0.0f;

    // snow bucket
    const float qs_out = fminf(ss, sm);
    ss = ss - qs_out + sn;
    const float qsp = qs_out + lp;
    const float qf_in = fmaxf(0.0f, qsp - f_thr);
    const float qu_in = fminf(qsp, f_thr);

    // fast reservoir
    sf += qf_in;
    const float qf_out = sf * fast_rcp(kf);
    sf -= qf_out;

    // unsaturated zone
    const float r   = su * inv_sumax;
    const float psi = fast_pow(r, beta);
    const float su_temp = fmaf(qu_in, 1.0f - psi, su);
    su = fminf(su_temp, sumax);
    const float qu_out = fmaf(qu_in, psi, fmaxf(0.0f, su_temp - sumax));
    const float ktheta = (su <= pwp) ? su * inv_sumax : 1.0f;
    const float ret = pet * KLU_C * ktheta;
    su = fmaxf(0.0f, su - ret);

    // interflow reservoir
    const float qi_in = qu_out * perc;
    si += qi_in;
    const float qi_out = si * fast_rcp(ki);
    si -= qi_out;

    // baseflow reservoir
    const float qb_in = qu_out - qi_in;   // qu_out * (1 - perc)
    sb += qb_in;
    const float qb_out = sb * fast_rcp(kb);
    sb -= qb_out;

    // mean over 16 models (butterfly within each 16-lane group; lowers to VALU permutes)
    float q = qf_out + qi_out + qb_out;
    q += __shfl_xor(q, 1, 32);
    q += __shfl_xor(q, 2, 32);
    q += __shfl_xor(q, 4, 32);
    q += __shfl_xor(q, 8, 32);
    if (mm == 0) out[outb + (size_t)t] = q * 0.0625f;
  };

  const int Tmain = T - KAHEAD;
  int t = 0;
#pragma unroll 1
  for (; t < Tmain; ++t) {
    WAIT_ASYNC_STEADY();          // guarantees slot t has landed (in-order completion)
    step(t);
    // refill: issue slot t + KAHEAD (== slot (t-1)&31, already consumed -> no alias)
    const unsigned slot = (unsigned)((t + KAHEAD) & (SLOTS - 1));
    const unsigned l0 = lds_wave + slot * (SLOT_F * 4u) + lane16;
    async_copy_slot(lstm, goff0, goff0 + db1, l0, l0 + 576u);
    goff0 += rowB;
  }
  WAIT_ASYNC_ZERO();              // drain remaining in-flight slots
#pragma unroll 1
  for (; t < T; ++t) step(t);
}

// ---------------- launch ----------------
extern "C" void kernel_launch(void* const* d_in, const int* in_sizes, int n_in,
                              void* d_out, int out_size, void* d_ws, size_t ws_size,
                              hipStream_t stream) {
  const float* xc   = (const float*)d_in[0];   // [B,T,3]
  const float* lstm = (const float*)d_in[1];   // [B,T,128]
  float*       out  = (float*)d_out;           // [B,T,1]
  (void)d_ws; (void)ws_size; (void)n_in; (void)out_size;

  const int B = 256;
  const int T = in_sizes[0] / (B * 3);         // 1460

  // 4096 chains = 128 wave32s; 8 waves/block -> 16 blocks, each wave owns 2 basins
  const int threads = WAVES_PER_BLOCK * 32;
  const int blocks  = (B * 16) / threads;      // B*M / 256 = 16
  hipLaunchKernelGGL(shm_scan_kernel, dim3(blocks), dim3(threads), 0, stream,
                     xc, lstm, out, T);
}